// GAT_824633721273
// MI455X (gfx1250) — compile-verified
//
#include <hip/hip_runtime.h>

#define N_NODES 100000
#define N_EDGES 800000
#define EPRIME  (N_EDGES + N_NODES)   // self-loops appended
#define HC      128
#define HEADS   4
#define HID     32
#define OUTC    40
#define OUTP    48                    // 40 padded to 3 WMMA col-tiles
#define NEG_SLOPE 0.2f
#define BN_EPS  1e-5f

typedef float v2f __attribute__((ext_vector_type(2)));
typedef float v8f __attribute__((ext_vector_type(8)));

// ---------------------------------------------------------------------------
// FP32 WMMA GEMM: Y[16x16 tile] = X[16x128] * W[128x16]; one wave per tile.
// A layout (16x4 f32): lane = (K>=2)*16 + M ; vgpr j holds K = 2*(lane>=16)+j
// B layout (4x16 f32): lane = (K>=2)*16 + N ; symmetric
// D layout (16x16 f32): vgpr r -> M = r + 8*(lane>=16), N = lane&15
// X row stride is always HC(=128). ldw/ldy parameterize the output width.
// ---------------------------------------------------------------------------
__global__ __launch_bounds__(32)
void gat_gemm_wmma(const float* __restrict__ X, const float* __restrict__ W,
                   float* __restrict__ Y, int ldw, int ldy) {
  const int lane = threadIdx.x;
  const int half = lane >> 4;        // 0 or 1
  const int lm   = lane & 15;
  const int row0 = blockIdx.x * 16;
  const int col0 = blockIdx.y * 16;

  const float* __restrict__ xrow = X + (size_t)(row0 + lm) * HC + half * 2;
  const float* __restrict__ wcol = W + (size_t)(half * 2) * ldw + col0 + lm;

  v8f acc = {};
#pragma unroll 4
  for (int k = 0; k < HC; k += 4) {
    v2f a, b;
    a.x = xrow[k];
    a.y = xrow[k + 1];
    b.x = wcol[(size_t)k * ldw];
    b.y = wcol[(size_t)(k + 1) * ldw];
    acc = __builtin_amdgcn_wmma_f32_16x16x4_f32(false, a, false, b,
                                                (short)0, acc, false, false);
  }
#pragma unroll
  for (int r = 0; r < 8; ++r) {
    int m = r + half * 8;
    Y[(size_t)(row0 + m) * ldy + col0 + lm] = acc[r];
  }
}

// ---------------------------------------------------------------------------
// Per-(node,head) attention coefficients: al = sum_c h[n,h,c] * a[h,c]
// ---------------------------------------------------------------------------
__global__ void gat_attn_coef(const float* __restrict__ h,
                              const float* __restrict__ a_src,
                              const float* __restrict__ a_dst,
                              float* __restrict__ als, float* __restrict__ ald,
                              int Hn, int C, int ldh) {
  int idx = blockIdx.x * blockDim.x + threadIdx.x;
  if (idx >= N_NODES * Hn) return;
  int n = idx / Hn, hh = idx - n * Hn;
  const float* __restrict__ hp = h + (size_t)n * ldh + hh * C;
  const float* __restrict__ asp = a_src + hh * C;
  const float* __restrict__ adp = a_dst + hh * C;
  float ss = 0.f, sd = 0.f;
  for (int c = 0; c < C; ++c) {
    float v = hp[c];
    ss = fmaf(v, asp[c], ss);
    sd = fmaf(v, adp[c], sd);
  }
  als[idx] = ss;
  ald[idx] = sd;
}

// ---------------------------------------------------------------------------
// order-preserving float<->uint key for atomic max (handles negatives)
// ---------------------------------------------------------------------------
__device__ __forceinline__ unsigned fkey(float f) {
  unsigned u = __float_as_uint(f);
  return (u & 0x80000000u) ? ~u : (u | 0x80000000u);
}
__device__ __forceinline__ float fdec(unsigned k) {
  unsigned u = (k & 0x80000000u) ? (k ^ 0x80000000u) : ~k;
  return __uint_as_float(u);
}
__device__ __forceinline__ void edge_sd(const int* __restrict__ ei, int i,
                                        int& s, int& d) {
  if (i < N_EDGES) { s = ei[i]; d = ei[N_EDGES + i]; }
  else             { s = d = i - N_EDGES; }           // self-loop
}
__device__ __forceinline__ float lrelu(float e) {
  return e > 0.f ? e : NEG_SLOPE * e;
}

__global__ void gat_zero(float* __restrict__ p, int n) {
  int i = blockIdx.x * blockDim.x + threadIdx.x;
  if (i < n) p[i] = 0.f;
}
__global__ void gat_init_mkey(unsigned* __restrict__ p, int n) {
  int i = blockIdx.x * blockDim.x + threadIdx.x;
  if (i < n) p[i] = fkey(-__builtin_inff());
}

// pass 1: segment max of leaky_relu(al_src[src]+al_dst[dst]) over dst
__global__ void gat_edge_max(const int* __restrict__ ei,
                             const float* __restrict__ als,
                             const float* __restrict__ ald,
                             unsigned* __restrict__ mkey, int Hn) {
  int idx = blockIdx.x * blockDim.x + threadIdx.x;
  if (idx >= EPRIME * Hn) return;
  int i = idx / Hn, hh = idx - i * Hn;
  int s, d; edge_sd(ei, i, s, d);
  float e = lrelu(als[s * Hn + hh] + ald[d * Hn + hh]);
  atomicMax(&mkey[d * Hn + hh], fkey(e));
}

// pass 2: segment sum of exp(e - m[dst])
__global__ void gat_edge_sum(const int* __restrict__ ei,
                             const float* __restrict__ als,
                             const float* __restrict__ ald,
                             const unsigned* __restrict__ mkey,
                             float* __restrict__ ssum, int Hn) {
  int idx = blockIdx.x * blockDim.x + threadIdx.x;
  if (idx >= EPRIME * Hn) return;
  int i = idx / Hn, hh = idx - i * Hn;
  int s, d; edge_sd(ei, i, s, d);
  float e = lrelu(als[s * Hn + hh] + ald[d * Hn + hh]);
  float m = fdec(mkey[d * Hn + hh]);
  atomicAdd(&ssum[d * Hn + hh], expf(e - m));
}

// pass 3 (layers 1/2): wave per edge, lane covers 4 channels; head = lane>>3
__global__ void gat_edge_agg128(const int* __restrict__ ei,
                                const float* __restrict__ h,
                                const float* __restrict__ als,
                                const float* __restrict__ ald,
                                const unsigned* __restrict__ mkey,
                                const float* __restrict__ ssum,
                                float* __restrict__ agg) {
  int wave = (blockIdx.x * blockDim.x + threadIdx.x) >> 5;
  int lane = threadIdx.x & 31;
  if (wave >= EPRIME) return;
  int s, d; edge_sd(ei, wave, s, d);
  int hh = lane >> 3;
  float e = lrelu(als[s * HEADS + hh] + ald[d * HEADS + hh]);
  float m = fdec(mkey[d * HEADS + hh]);
  float alpha = expf(e - m) / (ssum[d * HEADS + hh] + 1e-16f);
  const float4 hv = *(const float4*)(h + (size_t)s * HC + lane * 4);
  float* op = agg + (size_t)d * HC + lane * 4;
  atomicAdd(op + 0, alpha * hv.x);
  atomicAdd(op + 1, alpha * hv.y);
  atomicAdd(op + 2, alpha * hv.z);
  atomicAdd(op + 3, alpha * hv.w);
}

// pass 3 (layer 3): wave per edge, 40 channels, single head
__global__ void gat_edge_agg40(const int* __restrict__ ei,
                               const float* __restrict__ h,
                               const float* __restrict__ als,
                               const float* __restrict__ ald,
                               const unsigned* __restrict__ mkey,
                               const float* __restrict__ ssum,
                               float* __restrict__ agg) {
  int wave = (blockIdx.x * blockDim.x + threadIdx.x) >> 5;
  int lane = threadIdx.x & 31;
  if (wave >= EPRIME) return;
  int s, d; edge_sd(ei, wave, s, d);
  float e = lrelu(als[s] + ald[d]);
  float m = fdec(mkey[d]);
  float alpha = expf(e - m) / (ssum[d] + 1e-16f);
  const float* __restrict__ hp = h + (size_t)s * OUTP;
  float* op = agg + (size_t)d * OUTC;
  atomicAdd(op + lane, alpha * hp[lane]);
  if (lane < OUTC - 32) atomicAdd(op + 32 + lane, alpha * hp[32 + lane]);
}

// bias + batchnorm(eval) + relu, in place over [N, HC]
__global__ void gat_bias_bn_relu(float* __restrict__ h,
                                 const float* __restrict__ b,
                                 const float* __restrict__ g,
                                 const float* __restrict__ be,
                                 const float* __restrict__ rm,
                                 const float* __restrict__ rv) {
  int idx = blockIdx.x * blockDim.x + threadIdx.x;
  if (idx >= N_NODES * HC) return;
  int j = idx & (HC - 1);
  float v = h[idx] + b[j];
  v = (v - rm[j]) * rsqrtf(rv[j] + BN_EPS) * g[j] + be[j];
  h[idx] = v > 0.f ? v : 0.f;
}

// pad W3 [128,40] -> [128,48] with zeros
__global__ void gat_pad_w3(const float* __restrict__ W3, float* __restrict__ Wp) {
  int idx = blockIdx.x * blockDim.x + threadIdx.x;
  if (idx >= HC * OUTP) return;
  int k = idx / OUTP, c = idx - k * OUTP;
  Wp[idx] = (c < OUTC) ? W3[k * OUTC + c] : 0.f;
}

__global__ void gat_final_bias(const float* __restrict__ agg,
                               const float* __restrict__ b3,
                               float* __restrict__ out) {
  int idx = blockIdx.x * blockDim.x + threadIdx.x;
  if (idx >= N_NODES * OUTC) return;
  int c = idx % OUTC;
  out[idx] = agg[idx] + b3[c];
}

// ---------------------------------------------------------------------------
extern "C" void kernel_launch(void* const* d_in, const int* in_sizes, int n_in,
                              void* d_out, int out_size, void* d_ws, size_t ws_size,
                              hipStream_t stream) {
  const float* x   = (const float*)d_in[0];
  const int*   ei  = (const int*)d_in[1];
  const float* W1  = (const float*)d_in[2];
  const float* as1 = (const float*)d_in[3];
  const float* ad1 = (const float*)d_in[4];
  const float* b1  = (const float*)d_in[5];
  const float* g1  = (const float*)d_in[6];
  const float* be1 = (const float*)d_in[7];
  const float* rm1 = (const float*)d_in[8];
  const float* rv1 = (const float*)d_in[9];
  const float* W2  = (const float*)d_in[10];
  const float* as2 = (const float*)d_in[11];
  const float* ad2 = (const float*)d_in[12];
  const float* b2  = (const float*)d_in[13];
  const float* g2  = (const float*)d_in[14];
  const float* be2 = (const float*)d_in[15];
  const float* rm2 = (const float*)d_in[16];
  const float* rv2 = (const float*)d_in[17];
  const float* W3  = (const float*)d_in[18];
  const float* as3 = (const float*)d_in[19];
  const float* ad3 = (const float*)d_in[20];
  const float* b3  = (const float*)d_in[21];
  float* out = (float*)d_out;

  float* ws = (float*)d_ws;
  size_t o = 0;
  float*    A    = ws + o; o += (size_t)N_NODES * HC;     // gemm out / h
  float*    B    = ws + o; o += (size_t)N_NODES * HC;     // aggregate / h_next
  float*    H3   = ws + o; o += (size_t)N_NODES * OUTP;   // layer-3 gemm out
  float*    AGG3 = ws + o; o += (size_t)N_NODES * OUTC;
  float*    ALS  = ws + o; o += (size_t)N_NODES * HEADS;
  float*    ALD  = ws + o; o += (size_t)N_NODES * HEADS;
  unsigned* MKEY = (unsigned*)(ws + o); o += (size_t)N_NODES * HEADS;
  float*    SSUM = ws + o; o += (size_t)N_NODES * HEADS;
  float*    WPAD = ws + o; o += (size_t)HC * OUTP;
  (void)ws_size; (void)n_in; (void)in_sizes; (void)out_size;

  const int T = 256;
  const dim3 gemmGrid(N_NODES / 16, HC / 16);
  const dim3 gemm3Grid(N_NODES / 16, OUTP / 16);
  const int nhcBlk  = (N_NODES * HC + T - 1) / T;
  const int nh4Blk  = (N_NODES * HEADS + T - 1) / T;
  const int nh1Blk  = (N_NODES + T - 1) / T;
  const int eh4Blk  = (EPRIME * HEADS + T - 1) / T;
  const int eh1Blk  = (EPRIME + T - 1) / T;
  const int waveBlk = (EPRIME + 7) / 8;                 // 8 waves/block

  // ---------------- layer 1 ----------------
  gat_gemm_wmma<<<gemmGrid, 32, 0, stream>>>(x, W1, A, HC, HC);
  gat_attn_coef<<<nh4Blk, T, 0, stream>>>(A, as1, ad1, ALS, ALD, HEADS, HID, HC);
  gat_zero<<<nhcBlk, T, 0, stream>>>(B, N_NODES * HC);
  gat_init_mkey<<<nh4Blk, T, 0, stream>>>(MKEY, N_NODES * HEADS);
  gat_zero<<<nh4Blk, T, 0, stream>>>(SSUM, N_NODES * HEADS);
  gat_edge_max<<<eh4Blk, T, 0, stream>>>(ei, ALS, ALD, MKEY, HEADS);
  gat_edge_sum<<<eh4Blk, T, 0, stream>>>(ei, ALS, ALD, MKEY, SSUM, HEADS);
  gat_edge_agg128<<<waveBlk, T, 0, stream>>>(ei, A, ALS, ALD, MKEY, SSUM, B);
  gat_bias_bn_relu<<<nhcBlk, T, 0, stream>>>(B, b1, g1, be1, rm1, rv1);

  // ---------------- layer 2 ----------------
  gat_gemm_wmma<<<gemmGrid, 32, 0, stream>>>(B, W2, A, HC, HC);
  gat_attn_coef<<<nh4Blk, T, 0, stream>>>(A, as2, ad2, ALS, ALD, HEADS, HID, HC);
  gat_zero<<<nhcBlk, T, 0, stream>>>(B, N_NODES * HC);
  gat_init_mkey<<<nh4Blk, T, 0, stream>>>(MKEY, N_NODES * HEADS);
  gat_zero<<<nh4Blk, T, 0, stream>>>(SSUM, N_NODES * HEADS);
  gat_edge_max<<<eh4Blk, T, 0, stream>>>(ei, ALS, ALD, MKEY, HEADS);
  gat_edge_sum<<<eh4Blk, T, 0, stream>>>(ei, ALS, ALD, MKEY, SSUM, HEADS);
  gat_edge_agg128<<<waveBlk, T, 0, stream>>>(ei, A, ALS, ALD, MKEY, SSUM, B);
  gat_bias_bn_relu<<<nhcBlk, T, 0, stream>>>(B, b2, g2, be2, rm2, rv2);

  // ---------------- layer 3 ----------------
  gat_pad_w3<<<(HC * OUTP + T - 1) / T, T, 0, stream>>>(W3, WPAD);
  gat_gemm_wmma<<<gemm3Grid, 32, 0, stream>>>(B, WPAD, H3, OUTP, OUTP);
  gat_attn_coef<<<nh1Blk, T, 0, stream>>>(H3, as3, ad3, ALS, ALD, 1, OUTC, OUTP);
  gat_zero<<<(N_NODES * OUTC + T - 1) / T, T, 0, stream>>>(AGG3, N_NODES * OUTC);
  gat_init_mkey<<<nh1Blk, T, 0, stream>>>(MKEY, N_NODES);
  gat_zero<<<nh1Blk, T, 0, stream>>>(SSUM, N_NODES);
  gat_edge_max<<<eh1Blk, T, 0, stream>>>(ei, ALS, ALD, MKEY, 1);
  gat_edge_sum<<<eh1Blk, T, 0, stream>>>(ei, ALS, ALD, MKEY, SSUM, 1);
  gat_edge_agg40<<<waveBlk, T, 0, stream>>>(ei, H3, ALS, ALD, MKEY, SSUM, AGG3);
  gat_final_bias<<<(N_NODES * OUTC + T - 1) / T, T, 0, stream>>>(AGG3, b3, out);
}